// Pruner_1881195676112
// MI455X (gfx1250) — compile-verified
//
#include <hip/hip_runtime.h>

typedef __attribute__((ext_vector_type(2))) float v2f;
typedef __attribute__((ext_vector_type(8))) float v8f;

static constexpr int B_ = 16, N_ = 4096, E_ = 512, K_ = 1024;
#define NEGV (-1e20f)

// ---------------------------------------------------------------------------
// Kernel 1: scores = emb @ W + b, masked to NEGV.
// One wave = one 16-row tile. V_WMMA_F32_16X16X4_F32 with B = W chunk
// replicated across all 16 columns -> every column of D is the dot product.
// A layout (ISA 7.12.2, 32-bit A 16x4): lanes 0-15 hold K=0,1; lanes 16-31
// hold K=2,3 for row M=lane&15.  B rows are lane-constant, so the per-lane
// feed is the same (W[k+2h], W[k+2h+1]).
// ---------------------------------------------------------------------------
__global__ __launch_bounds__(256) void score_kernel(
    const float* __restrict__ emb, const float* __restrict__ W,
    const float* __restrict__ bias, const int* __restrict__ mask,
    float* __restrict__ scores)
{
  const int wave = threadIdx.x >> 5;
  const int lane = threadIdx.x & 31;
  const int tile = blockIdx.x * 8 + wave;       // 16-row tile index
  const int tilesPerBatch = N_ / 16;
  const int b    = tile / tilesPerBatch;
  const int row0 = (tile % tilesPerBatch) * 16;
  const int half = lane >> 4;                   // 0: K=0,1   1: K=2,3
  const int m    = lane & 15;                   // row within tile
  const float* rowp = emb + ((size_t)b * N_ + (row0 + m)) * E_;

  v8f acc = {};
#pragma unroll 4
  for (int k = 0; k < E_; k += 4) {
    v2f a, w2;
    a.x  = rowp[k + 2 * half];
    a.y  = rowp[k + 2 * half + 1];
    w2.x = W[k + 2 * half];
    w2.y = W[k + 2 * half + 1];
    acc = __builtin_amdgcn_wmma_f32_16x16x4_f32(
        false, a, false, w2, (short)0, acc, false, false);
  }

  // C/D layout: VGPR j, lane 0 -> (M=j, N=0); lane 16 -> (M=8+j, N=0).
  if (m == 0) {
    const float bv = bias[0];
#pragma unroll
    for (int j = 0; j < 8; ++j) {
      int r   = row0 + half * 8 + j;
      float s = acc[j] + bv;
      s = (mask[b * N_ + r] > 0) ? s : NEGV;
      scores[b * N_ + r] = s;
    }
  }
}

// ---------------------------------------------------------------------------
// Kernel 2: per-row top-K.  One 1024-thread workgroup per batch row.
// Key = (flip(score) << 32) | (0xFFFFFFFF - idx): descending bitonic sort
// reproduces lax.top_k order (ties -> smaller index first).
// fill = max index over top-K >= every kept index, so final result is
// sort(kept) padded with fill.
// ---------------------------------------------------------------------------
__global__ __launch_bounds__(1024) void topk_kernel(
    const float* __restrict__ scores, const int* __restrict__ mask,
    const int* __restrict__ ntk_arr, int* __restrict__ idx_out,
    int* __restrict__ mask_out, float* __restrict__ score_out)
{
  __shared__ unsigned long long keys[N_];   // 32 KB
  __shared__ int s_idx[K_];                 // 4 KB
  __shared__ int s_fill;

  const int b   = blockIdx.x;
  const int tid = threadIdx.x;

  for (int i = tid; i < N_; i += 1024) {
    unsigned u = __float_as_uint(scores[b * N_ + i]);
    u = (u & 0x80000000u) ? ~u : (u | 0x80000000u);   // monotonic map
    keys[i] = ((unsigned long long)u << 32) |
              (unsigned long long)(0xFFFFFFFFu - (unsigned)i);
  }
  __syncthreads();

  // Bitonic sort, descending, 4096 keys with 1024 threads (4 comparators/thr)
  for (unsigned size = 2; size <= (unsigned)N_; size <<= 1) {
    for (unsigned stride = size >> 1; stride > 0; stride >>= 1) {
      for (unsigned t = (unsigned)tid; t < (unsigned)N_; t += 1024u) {
        unsigned p = t ^ stride;
        if (p > t) {
          unsigned long long a = keys[t], c = keys[p];
          bool descBlock = ((t & size) == 0);
          if (descBlock ? (a < c) : (a > c)) { keys[t] = c; keys[p] = a; }
        }
      }
      __syncthreads();
    }
  }

  const int ntk = ntk_arr[b];
  int myidx = (int)(0xFFFFFFFFu - (unsigned)(keys[tid] & 0xFFFFFFFFull));
  s_idx[tid] = myidx;
  if (tid == 0) s_fill = 0;
  __syncthreads();
  atomicMax(&s_fill, myidx);
  __syncthreads();

  int* ik = (int*)keys;   // reuse LDS for the 1024-int ascending sort
  ik[tid] = (tid < ntk) ? s_idx[tid] : 0x7FFFFFFF;
  __syncthreads();

  for (unsigned size = 2; size <= (unsigned)K_; size <<= 1) {
    for (unsigned stride = size >> 1; stride > 0; stride >>= 1) {
      unsigned p = (unsigned)tid ^ stride;
      if (p > (unsigned)tid) {
        int a = ik[tid], c = ik[p];
        bool asc = (((unsigned)tid & size) == 0);
        if (asc ? (a > c) : (a < c)) { ik[tid] = c; ik[p] = a; }
      }
      __syncthreads();
    }
  }

  const int fin = (tid < ntk) ? ik[tid] : s_fill;
  idx_out[b * K_ + tid]   = fin;
  mask_out[b * K_ + tid]  = (tid < ntk && mask[b * N_ + fin] > 0) ? 1 : 0;
  score_out[b * K_ + tid] = scores[b * N_ + fin];
}

// ---------------------------------------------------------------------------
// Kernel 3: embedding gather [B,K,E].  One block per (b,k); 128 lanes x
// float4 = 512 floats (b128 loads/stores; source rows should be L2-resident).
// ---------------------------------------------------------------------------
__global__ __launch_bounds__(128) void gather_kernel(
    const float* __restrict__ emb, const int* __restrict__ idx,
    float* __restrict__ out)
{
  const int bk  = blockIdx.x;          // b*K + slot
  const int b   = bk >> 10;            // K_ == 1024
  const int src = idx[bk];
  const float4* s = (const float4*)(emb + ((size_t)b * N_ + src) * E_);
  float4*       d = (float4*)(out + (size_t)bk * E_);
  d[threadIdx.x] = s[threadIdx.x];
}

extern "C" void kernel_launch(void* const* d_in, const int* in_sizes, int n_in,
                              void* d_out, int out_size, void* d_ws, size_t ws_size,
                              hipStream_t stream)
{
  const float* emb  = (const float*)d_in[0];
  const float* W    = (const float*)d_in[1];
  const float* bias = (const float*)d_in[2];
  const int*   mask = (const int*)d_in[3];
  const int*   ntk  = (const int*)d_in[4];

  float* scores = (float*)d_ws;                       // 16*4096*4 = 256 KB

  char*  outc      = (char*)d_out;
  float* emb_out   = (float*)outc;                                        // [B,K,E]
  int*   mask_out  = (int*)(outc + (size_t)B_ * K_ * E_ * 4);             // [B,K]
  int*   idx_out   = (int*)(outc + ((size_t)B_ * K_ * E_ + (size_t)B_ * K_) * 4);
  float* score_out = (float*)(outc + ((size_t)B_ * K_ * E_ + 2 * (size_t)B_ * K_) * 4);

  score_kernel <<<(B_ * (N_ / 16)) / 8, 256, 0, stream>>>(emb, W, bias, mask, scores);
  topk_kernel  <<<B_, 1024, 0, stream>>>(scores, mask, ntk, idx_out, mask_out, score_out);
  gather_kernel<<<B_ * K_, 128, 0, stream>>>(emb, idx_out, emb_out);
}